// GaussianModel_11218454577222
// MI455X (gfx1250) — compile-verified
//
#include <hip/hip_runtime.h>
#include <stdint.h>

#define BLOCK 256   // 8 wave32 waves per block; 4,000,000 % 256 == 0 -> all blocks full

// Low 32 bits of a flat pointer to __shared__ memory == LDS byte address
// (LDS aperture lives entirely in addr[63:32]).
__device__ __forceinline__ uint32_t lds_off(const void* p) {
    return (uint32_t)(uintptr_t)p;
}

// Async DMA: 16 B global -> LDS per enabled lane (ASYNCcnt).
__device__ __forceinline__ void async_gl2lds_b128(uint32_t lds, const void* g) {
    asm volatile("global_load_async_to_lds_b128 %0, %1, off"
                 :: "v"(lds), "v"(g) : "memory");
}
// Async DMA: 16 B LDS -> global per enabled lane (ASYNCcnt).
__device__ __forceinline__ void async_lds2gl_b128(void* g, uint32_t lds) {
    asm volatile("global_store_async_from_lds_b128 %0, %1, off"
                 :: "v"(g), "v"(lds) : "memory");
}
__device__ __forceinline__ void wait_async0() {
    asm volatile("s_wait_asynccnt 0" ::: "memory");
}

__global__ __launch_bounds__(BLOCK) void gauss_cov_kernel(
    const float4* __restrict__ rot,   // (N,4) w,x,y,z
    const float*  __restrict__ scal,  // (N,3)
    float*        __restrict__ out,   // (N,3,3)
    int n)
{
    __shared__ float s_scal[BLOCK * 3];  // 3 KB  (async-DMA landing zone)
    __shared__ float s_cov [BLOCK * 9];  // 9 KB  (store staging for coalesced b128 DMA)

    const int  tid  = threadIdx.x;
    const long base = (long)blockIdx.x * BLOCK;
    const bool full = (base + BLOCK) <= (long)n;

    // --- kick off scaling-chunk DMA first: 3*BLOCK floats = 3072 B = 192 x b128 ---
    if (full) {
        if (tid < (BLOCK * 3) / 4) {
            const char* g = (const char*)(scal + base * 3) + 16 * tid;
            async_gl2lds_b128(lds_off(s_scal) + 16u * (uint32_t)tid, g);
        }
    } else {
        const long i = base + tid;
        if (i < n) {
            s_scal[3 * tid + 0] = scal[3 * i + 0];
            s_scal[3 * tid + 1] = scal[3 * i + 1];
            s_scal[3 * tid + 2] = scal[3 * i + 2];
        }
    }

    // --- overlap: quaternion load (coalesced b128) + rotation matrix VALU ---
    const long i = base + tid;
    float r00=0,r01=0,r02=0,r10=0,r11=0,r12=0,r20=0,r21=0,r22=0;
    if (i < n) {
        float4 q = rot[i];
        float w = q.x, x = q.y, y = q.z, z = q.w;
        float inv = rsqrtf(w*w + x*x + y*y + z*z);
        w *= inv; x *= inv; y *= inv; z *= inv;
        float xx=x*x, yy=y*y, zz=z*z;
        float xy=x*y, xz=x*z, yz=y*z;
        float wx=w*x, wy=w*y, wz=w*z;
        r00 = 1.f - 2.f*(yy + zz); r01 = 2.f*(xy - wz);       r02 = 2.f*(xz + wy);
        r10 = 2.f*(xy + wz);       r11 = 1.f - 2.f*(xx + zz); r12 = 2.f*(yz - wx);
        r20 = 2.f*(xz - wy);       r21 = 2.f*(yz + wx);       r22 = 1.f - 2.f*(xx + yy);
    }

    if (full) wait_async0();   // my wave's DMA landed
    __syncthreads();           // everyone's DMA landed

    // --- cov = R diag(s^2) R^T ; s_k^2 = exp(2*raw_k). 6 unique entries. ---
    if (i < n) {
        float e0 = expf(2.f * s_scal[3*tid + 0]);
        float e1 = expf(2.f * s_scal[3*tid + 1]);
        float e2 = expf(2.f * s_scal[3*tid + 2]);
        float c00 = r00*r00*e0 + r01*r01*e1 + r02*r02*e2;
        float c01 = r00*r10*e0 + r01*r11*e1 + r02*r12*e2;
        float c02 = r00*r20*e0 + r01*r21*e1 + r02*r22*e2;
        float c11 = r10*r10*e0 + r11*r11*e1 + r12*r12*e2;
        float c12 = r10*r20*e0 + r11*r21*e1 + r12*r22*e2;
        float c22 = r20*r20*e0 + r21*r21*e1 + r22*r22*e2;
        float* o = &s_cov[9 * tid];   // stride 9 words, gcd(9,64)=1 -> bank-conflict free
        o[0]=c00; o[1]=c01; o[2]=c02;
        o[3]=c01; o[4]=c11; o[5]=c12;
        o[6]=c02; o[7]=c12; o[8]=c22;
    }

    __syncthreads();

    // --- writeback: 9*BLOCK floats = 9216 B = 576 x b128 async LDS->global DMA ---
    float* gout = out + base * 9;
    if (full) {
        const uint32_t l0 = lds_off(s_cov);
        #pragma unroll
        for (int k = 0; k < 2; ++k) {
            int j = k * BLOCK + tid;
            async_lds2gl_b128((char*)gout + 16 * j, l0 + 16u * (uint32_t)j);
        }
        if (tid < (9 * BLOCK) / 4 - 2 * BLOCK) {   // remaining 64 transfers
            int j = 2 * BLOCK + tid;
            async_lds2gl_b128((char*)gout + 16 * j, l0 + 16u * (uint32_t)j);
        }
        // Fire-and-forget: ASYNCcnt drained by implicit S_WAIT_IDLE at S_ENDPGM;
        // LDS stays allocated until every wave in the workgroup has ended.
    } else {
        int nf = (int)(n - base) * 9;
        for (int k = tid; k < nf; k += BLOCK) gout[k] = s_cov[k];
    }
}

extern "C" void kernel_launch(void* const* d_in, const int* in_sizes, int n_in,
                              void* d_out, int out_size, void* d_ws, size_t ws_size,
                              hipStream_t stream) {
    const float4* rot  = (const float4*)d_in[0];
    const float*  scal = (const float*)d_in[1];
    float*        out  = (float*)d_out;
    const int n = in_sizes[0] / 4;          // rotation_raw is (N,4)
    const int blocks = (n + BLOCK - 1) / BLOCK;
    if (blocks > 0) {
        gauss_cov_kernel<<<dim3(blocks), dim3(BLOCK), 0, stream>>>(rot, scal, out, n);
    }
}